// LightGCN_65506841198663
// MI455X (gfx1250) — compile-verified
//
#include <hip/hip_runtime.h>

#define DIM 64
#define DIM4 (DIM / 4)   // float4 chunks per node row

// ---------------------------------------------------------------------------
// SpMM scatter: for each edge (r, c): next[r, :] += scale * cur[c, :]
// 16 threads per edge (half a wave32), each thread owns 4 contiguous floats:
//   - cur[c*64 + 4t .. +3] loaded as one global_load_b128 (256B/edge, coalesced)
//   - 4x global_atomic_add_f32 (no return -> STOREcnt only) into next[r*64+4t..]
// Edge-index loads are non-temporal (streamed once per hop; keep the two
// 38.4MB node buffers resident in the 192MB L2 instead), with a
// global_prefetch_b8 run-ahead on the edge stream.
// Placed first in the file so the disasm snippet shows its lowering.
// ---------------------------------------------------------------------------
__global__ void lightgcn_spmm_scatter(float* __restrict__ next,
                                      const float* __restrict__ cur,
                                      const int* __restrict__ rows,
                                      const int* __restrict__ cols,
                                      int n_edges, float scale) {
    long gid = (long)blockIdx.x * blockDim.x + threadIdx.x;
    int e = (int)(gid >> 4);        // edge index
    int t = (int)(gid & 15);        // float4 chunk within the 64-dim row
    if (e >= n_edges) return;

    // Run-ahead prefetch of the edge stream (gfx1250 global_prefetch_b8).
    if (t == 0) {
        int pe = e + 8192;
        if (pe < n_edges) {
            __builtin_prefetch(&rows[pe], 0, 1);
            __builtin_prefetch(&cols[pe], 0, 1);
        }
    }

    // Non-temporal: edge lists are touched once per hop; don't pollute L2.
    int r = __builtin_nontemporal_load(&rows[e]);
    int c = __builtin_nontemporal_load(&cols[e]);

    float4 v = *(const float4*)(cur + (long)c * DIM + (t << 2));
    v.x *= scale; v.y *= scale; v.z *= scale; v.w *= scale;
    float* dst = next + (long)r * DIM + (t << 2);
    // Hardware f32 atomics (L2-side RMW), no-return form.
    unsafeAtomicAdd(dst + 0, v.x);
    unsafeAtomicAdd(dst + 1, v.y);
    unsafeAtomicAdd(dst + 2, v.z);
    unsafeAtomicAdd(dst + 3, v.w);
}

// ---------------------------------------------------------------------------
// dst[node] = scale * concat(user_emb, item_emb)[node]  (float4 stream)
// ---------------------------------------------------------------------------
__global__ void lightgcn_init_emb(float* __restrict__ dst,
                                  const float* __restrict__ user_emb,
                                  const float* __restrict__ item_emb,
                                  int n_users, int total4, float scale) {
    int i = blockIdx.x * blockDim.x + threadIdx.x;   // float4 index
    if (i >= total4) return;
    int node = i >> 4;              // DIM4 == 16 -> /16
    float4 v;
    if (node < n_users) {
        v = ((const float4*)user_emb)[i];
    } else {
        v = ((const float4*)item_emb)[i - n_users * DIM4];
    }
    v.x *= scale; v.y *= scale; v.z *= scale; v.w *= scale;
    ((float4*)dst)[i] = v;
}

extern "C" void kernel_launch(void* const* d_in, const int* in_sizes, int n_in,
                              void* d_out, int out_size, void* d_ws, size_t ws_size,
                              hipStream_t stream) {
    const float* user_emb = (const float*)d_in[0];
    const float* item_emb = (const float*)d_in[1];
    const int*   rows     = (const int*)d_in[2];
    const int*   cols     = (const int*)d_in[3];
    float* out = (float*)d_out;

    const int n_users = in_sizes[0] / DIM;
    const int n_items = in_sizes[1] / DIM;
    const int n_nodes = n_users + n_items;
    const int n_edges = in_sizes[2];
    const int total4  = n_nodes * DIM4;          // float4 elements per buffer

    // Ping-pong buffers: bufA = scratch (38.4 MB), bufB = d_out.
    // Hop parity: A -> B -> A -> B, so hop 2 lands in d_out.
    float* bufA = (float*)d_ws;
    float* bufB = out;

    const int BLK = 256;
    const int grid_elem = (total4 + BLK - 1) / BLK;
    const long scatter_threads = (long)n_edges * 16;
    const int grid_sc = (int)((scatter_threads + BLK - 1) / BLK);

    // Horner: out = (emb + A(emb + A(emb + A*emb)))/4.
    // The 1/4 is folded into the LAST hop (init with 0.25*emb, scatter adds
    // 0.25*cur), eliminating a separate 77MB scale pass over d_out.
    lightgcn_init_emb<<<grid_elem, BLK, 0, stream>>>(bufA, user_emb, item_emb,
                                                     n_users, total4, 1.0f);
    float* cur = bufA;
    float* nxt = bufB;
    for (int hop = 0; hop < 3; ++hop) {
        const float s = (hop == 2) ? 0.25f : 1.0f;
        lightgcn_init_emb<<<grid_elem, BLK, 0, stream>>>(nxt, user_emb, item_emb,
                                                         n_users, total4, s);
        lightgcn_spmm_scatter<<<grid_sc, BLK, 0, stream>>>(nxt, cur, rows, cols,
                                                           n_edges, s);
        float* tmp = cur; cur = nxt; nxt = tmp;
    }
    // cur == bufB == d_out: done.
}